// GraphNeuralNetwork_23502061043716
// MI455X (gfx1250) — compile-verified
//
#include <hip/hip_runtime.h>
#include <math.h>

#define N_HEADS 4
#define CPH 32      // channels per head
#define HIDDIM 128
#define F_INDIM 64
#define NGRAPH 64
#define BN_EPS 1e-5f

typedef float v2f __attribute__((ext_vector_type(2)));
typedef float v8f __attribute__((ext_vector_type(8)));

static __device__ __forceinline__ void atomicMaxF(float* addr, float v) {
  // standard bit-punning float max: ints order positives, reversed uints order negatives
  if (v >= 0.0f)
    atomicMax((int*)addr, __float_as_int(v));
  else
    atomicMin((unsigned int*)addr, __float_as_uint(v));
}

// ---------------------------------------------------------------------------
// C[M,N] = A[M,K] @ B[K,N] (+bias[N]), all row-major, M%16==0, N%16==0, K%4==0
// One wave computes one 16x16 tile with V_WMMA_F32_16X16X4_F32.
// A 16x4 f32 operand layout: lanes 0-15 -> M=lane, VGPR0/1 = K0/K1;
//                            lanes 16-31 -> M=lane-16, VGPR0/1 = K2/K3.
// C/D layout: VGPR r -> (M = r + 8*(lane>=16), N = lane&15).
// ---------------------------------------------------------------------------
__global__ void gemm_wmma_f32(const float* __restrict__ A,
                              const float* __restrict__ B,
                              const float* __restrict__ bias,
                              float* __restrict__ C,
                              int M, int Nn, int K) {
  const int lane    = threadIdx.x & 31;
  const int wave    = threadIdx.x >> 5;
  const int rowTile = blockIdx.x;
  const int colTile = blockIdx.y * 8 + wave;
  if (colTile * 16 >= Nn) return;          // wave-uniform guard (EXEC stays all-1)
  const int r    = lane & 15;
  const int half = lane >> 4;              // 0: K pair {0,1}, 1: K pair {2,3}
  const int row  = rowTile * 16 + r;
  const int col  = colTile * 16 + r;
  const float* ap = A + (long)row * K + 2 * half;
  const float* bp = B + (long)(2 * half) * Nn + col;
  v8f acc = {};
  for (int k = 0; k < K; k += 4) {
    v2f a, b;
    a.x = ap[k + 0];
    a.y = ap[k + 1];
    b.x = bp[(long)k * Nn];
    b.y = bp[(long)(k + 1) * Nn];
    acc = __builtin_amdgcn_wmma_f32_16x16x4_f32(false, a, false, b,
                                                (short)0, acc, false, false);
  }
  const int rbase = rowTile * 16 + half * 8;
  const float badd = bias ? bias[col] : 0.0f;
#pragma unroll
  for (int i = 0; i < 8; ++i)
    C[(long)(rbase + i) * Nn + col] = acc[i] + badd;
}

// --------------------------- utility fill ----------------------------------
__global__ void fillf(float* __restrict__ p, float v, long n) {
  long i = (long)blockIdx.x * blockDim.x + threadIdx.x;
  if (i < n) p[i] = v;
}

// ---------------- per-(node,head) attention logits -------------------------
__global__ void attn_prep(const float* __restrict__ h,
                          const float* __restrict__ a_src,
                          const float* __restrict__ a_dst,
                          float* __restrict__ as, float* __restrict__ ad, int n) {
  int i = blockIdx.x * blockDim.x + threadIdx.x;     // over N * H
  if (i >= n * N_HEADS) return;
  int node = i / N_HEADS, hd = i % N_HEADS;
  const float* hp  = h + (long)node * HIDDIM + hd * CPH;
  const float* asw = a_src + hd * CPH;
  const float* adw = a_dst + hd * CPH;
  float s = 0.f, d = 0.f;
#pragma unroll
  for (int c = 0; c < CPH; ++c) { float v = hp[c]; s += v * asw[c]; d += v * adw[c]; }
  as[i] = s;
  ad[i] = d;
}

// ---------------- edge pass 1: segment max of leaky-relu logits ------------
__global__ void edge_max(const int* __restrict__ src, const int* __restrict__ dst,
                         const float* __restrict__ as, const float* __restrict__ ad,
                         float* __restrict__ m, int nE, int nNodes) {
  int i = blockIdx.x * blockDim.x + threadIdx.x;     // over (E+N) * H
  if (i >= (nE + nNodes) * N_HEADS) return;
  int e = i / N_HEADS, hd = i % N_HEADS;
  int s_ = (e < nE) ? src[e] : (e - nE);             // self loops appended
  int d_ = (e < nE) ? dst[e] : (e - nE);
  float x = as[s_ * N_HEADS + hd] + ad[d_ * N_HEADS + hd];
  x = (x > 0.f) ? x : 0.2f * x;
  atomicMaxF(&m[d_ * N_HEADS + hd], x);
}

// ---------------- edge pass 2: exp(e - max) and segment sum ----------------
__global__ void edge_exp(const int* __restrict__ src, const int* __restrict__ dst,
                         const float* __restrict__ as, const float* __restrict__ ad,
                         const float* __restrict__ m, float* __restrict__ eexp,
                         float* __restrict__ ssum, int nE, int nNodes) {
  int i = blockIdx.x * blockDim.x + threadIdx.x;
  if (i >= (nE + nNodes) * N_HEADS) return;
  int e = i / N_HEADS, hd = i % N_HEADS;
  int s_ = (e < nE) ? src[e] : (e - nE);
  int d_ = (e < nE) ? dst[e] : (e - nE);
  float x = as[s_ * N_HEADS + hd] + ad[d_ * N_HEADS + hd];
  x = (x > 0.f) ? x : 0.2f * x;
  float v = expf(x - m[d_ * N_HEADS + hd]);
  eexp[i] = v;
  atomicAdd(&ssum[d_ * N_HEADS + hd], v);
}

// ---------------- edge pass 3: alpha-weighted aggregation ------------------
// one wave per edge; each lane handles 4 channels (float4 gather)
__global__ void edge_aggr(const int* __restrict__ src, const int* __restrict__ dst,
                          const float* __restrict__ h, const float* __restrict__ eexp,
                          const float* __restrict__ ssum, float* __restrict__ aggr,
                          int nE, int nNodes) {
  int lane = threadIdx.x & 31;
  int e = (int)(((long)blockIdx.x * blockDim.x + threadIdx.x) >> 5);
  if (e >= nE + nNodes) return;
  int s_ = (e < nE) ? src[e] : (e - nE);
  int d_ = (e < nE) ? dst[e] : (e - nE);
  int c0 = lane * 4;                                  // 0..124
  int hd = c0 >> 5;                                   // head for these channels
  float alpha = eexp[e * N_HEADS + hd] /
                (ssum[d_ * N_HEADS + hd] + 1e-16f);
  const float4 hv = *(const float4*)(h + (long)s_ * HIDDIM + c0);
  float* out = aggr + (long)d_ * HIDDIM + c0;
  atomicAdd(out + 0, hv.x * alpha);
  atomicAdd(out + 1, hv.y * alpha);
  atomicAdd(out + 2, hv.z * alpha);
  atomicAdd(out + 3, hv.w * alpha);
}

// ---------------- batch-norm statistics (sum / sumsq per channel) ----------
__global__ void bn_stats(const float* __restrict__ x, float* __restrict__ sum,
                         float* __restrict__ sumsq, int n) {
  const int ch = threadIdx.x;                         // blockDim.x == 128
  const int chunk = (n + gridDim.x - 1) / gridDim.x;
  const int n0 = blockIdx.x * chunk;
  const int n1 = (n0 + chunk < n) ? (n0 + chunk) : n;
  float s = 0.f, s2 = 0.f;
  for (int i = n0; i < n1; ++i) {
    float v = x[(long)i * HIDDIM + ch];
    s += v;
    s2 += v * v;
  }
  atomicAdd(&sum[ch], s);
  atomicAdd(&sumsq[ch], s2);
}

// ---------------- batch-norm apply + ReLU ----------------------------------
__global__ void bn_apply(const float* __restrict__ x, const float* __restrict__ sum,
                         const float* __restrict__ sumsq,
                         const float* __restrict__ gamma,
                         const float* __restrict__ beta,
                         float* __restrict__ y, int n) {
  long i = (long)blockIdx.x * blockDim.x + threadIdx.x;
  if (i >= (long)n * HIDDIM) return;
  int ch = (int)(i % HIDDIM);
  float inv_n = 1.0f / (float)n;
  float mu  = sum[ch] * inv_n;
  float var = sumsq[ch] * inv_n - mu * mu;
  float v = (x[i] - mu) * rsqrtf(var + BN_EPS) * gamma[ch] + beta[ch];
  y[i] = fmaxf(v, 0.0f);
}

// ---------------- per-graph mean/max pooling -------------------------------
__global__ void pool_scatter(const float* __restrict__ h, const int* __restrict__ batch,
                             float* __restrict__ msum, float* __restrict__ mmax,
                             float* __restrict__ cnt, int n) {
  long i = (long)blockIdx.x * blockDim.x + threadIdx.x;
  if (i >= (long)n * HIDDIM) return;
  int node = (int)(i / HIDDIM), ch = (int)(i % HIDDIM);
  int g = batch[node];
  float v = h[i];
  atomicAdd(&msum[g * HIDDIM + ch], v);
  atomicMaxF(&mmax[g * HIDDIM + ch], v);
  if (ch == 0) atomicAdd(&cnt[g], 1.0f);
}

__global__ void pool_finalize(const float* __restrict__ msum,
                              const float* __restrict__ mmax,
                              const float* __restrict__ cnt,
                              float* __restrict__ repr) {
  int i = blockIdx.x * blockDim.x + threadIdx.x;      // over G * HID
  if (i >= NGRAPH * HIDDIM) return;
  int g = i / HIDDIM, ch = i % HIDDIM;
  float c = fmaxf(cnt[g], 1.0f);
  repr[g * 2 * HIDDIM + ch]          = msum[i] / c;   // mean pool
  repr[g * 2 * HIDDIM + HIDDIM + ch] = mmax[i];       // max pool
}

// ===========================================================================
static inline int cdiv(long a, int b) { return (int)((a + b - 1) / b); }

extern "C" void kernel_launch(void* const* d_in, const int* in_sizes, int n_in,
                              void* d_out, int out_size, void* d_ws, size_t ws_size,
                              hipStream_t stream) {
  const float* x     = (const float*)d_in[0];
  const int*   ei    = (const int*)d_in[1];   // [2,E] flat: src row then dst row
  const int*   batch = (const int*)d_in[2];
  const int N    = in_sizes[2];
  const int E    = in_sizes[1] / 2;
  const int Etot = E + N;                     // with self loops

  // params in setup_inputs() insertion order:
  // per layer: W, a_src, a_dst, bias, gamma, beta ; then proj_W, proj_b
  const float *W[3], *a_src[3], *a_dst[3], *gamma[3], *beta[3];
  for (int l = 0; l < 3; ++l) {
    W[l]     = (const float*)d_in[3 + 6 * l + 0];
    a_src[l] = (const float*)d_in[3 + 6 * l + 1];
    a_dst[l] = (const float*)d_in[3 + 6 * l + 2];
    // bias (index +3) is mathematically a no-op: it cancels in batch-norm mean
    gamma[l] = (const float*)d_in[3 + 6 * l + 4];
    beta[l]  = (const float*)d_in[3 + 6 * l + 5];
  }
  const float* projW = (const float*)d_in[21];
  const float* projB = (const float*)d_in[22];

  // ----- workspace carve (256B aligned) -----
  char* wp = (char*)d_ws;
  auto carve = [&](size_t bytes) -> float* {
    char* r = wp;
    wp += (bytes + 255) & ~(size_t)255;
    return (float*)r;
  };
  float* hw   = carve((size_t)N * HIDDIM * 4);        // GEMM output h
  float* hcur = carve((size_t)N * HIDDIM * 4);        // layer output (BN+ReLU)
  float* aggr = carve((size_t)N * HIDDIM * 4);        // attention aggregation
  float* as   = carve((size_t)N * N_HEADS * 4);
  float* ad   = carve((size_t)N * N_HEADS * 4);
  float* m    = carve((size_t)N * N_HEADS * 4);
  float* ssum = carve((size_t)N * N_HEADS * 4);
  float* eexp = carve((size_t)Etot * N_HEADS * 4);
  float* stat = carve((size_t)2 * HIDDIM * 4);        // sum | sumsq
  float* psum = carve((size_t)NGRAPH * HIDDIM * 4);
  float* pmax = carve((size_t)NGRAPH * HIDDIM * 4);
  float* pcnt = carve((size_t)NGRAPH * 4);
  float* repr = carve((size_t)NGRAPH * 2 * HIDDIM * 4);

  const int* src = ei;
  const int* dst = ei + E;

  const float* in = x;
  int K = F_INDIM;
  for (int l = 0; l < 3; ++l) {
    // h = in @ W[l]   (WMMA fp32)
    dim3 ggrid(N / 16, 1);
    gemm_wmma_f32<<<ggrid, 256, 0, stream>>>(in, W[l], nullptr, hw, N, HIDDIM, K);

    // attention logits per (node, head)
    attn_prep<<<cdiv((long)N * N_HEADS, 256), 256, 0, stream>>>(hw, a_src[l], a_dst[l],
                                                                as, ad, N);
    // init segment buffers
    fillf<<<cdiv((long)N * N_HEADS, 256), 256, 0, stream>>>(m, -INFINITY, (long)N * N_HEADS);
    fillf<<<cdiv((long)N * N_HEADS, 256), 256, 0, stream>>>(ssum, 0.0f, (long)N * N_HEADS);
    fillf<<<cdiv((long)N * HIDDIM, 256), 256, 0, stream>>>(aggr, 0.0f, (long)N * HIDDIM);

    // scatter-softmax + aggregation over edges (+ self loops)
    edge_max<<<cdiv((long)Etot * N_HEADS, 256), 256, 0, stream>>>(src, dst, as, ad, m, E, N);
    edge_exp<<<cdiv((long)Etot * N_HEADS, 256), 256, 0, stream>>>(src, dst, as, ad, m,
                                                                  eexp, ssum, E, N);
    edge_aggr<<<cdiv((long)Etot * 32, 256), 256, 0, stream>>>(src, dst, hw, eexp, ssum,
                                                              aggr, E, N);

    // batch norm (+ReLU) -> hcur
    fillf<<<1, 2 * HIDDIM, 0, stream>>>(stat, 0.0f, 2 * HIDDIM);
    bn_stats<<<256, HIDDIM, 0, stream>>>(aggr, stat, stat + HIDDIM, N);
    bn_apply<<<cdiv((long)N * HIDDIM, 256), 256, 0, stream>>>(aggr, stat, stat + HIDDIM,
                                                              gamma[l], beta[l], hcur, N);
    in = hcur;
    K = HIDDIM;
  }

  // per-graph mean/max pooling
  fillf<<<cdiv((long)NGRAPH * HIDDIM, 256), 256, 0, stream>>>(psum, 0.0f, (long)NGRAPH * HIDDIM);
  fillf<<<cdiv((long)NGRAPH * HIDDIM, 256), 256, 0, stream>>>(pmax, -INFINITY, (long)NGRAPH * HIDDIM);
  fillf<<<1, NGRAPH, 0, stream>>>(pcnt, 0.0f, NGRAPH);
  pool_scatter<<<cdiv((long)N * HIDDIM, 256), 256, 0, stream>>>(hcur, batch, psum, pmax, pcnt, N);
  pool_finalize<<<cdiv((long)NGRAPH * HIDDIM, 256), 256, 0, stream>>>(psum, pmax, pcnt, repr);

  // final projection: out[G,128] = repr[G,256] @ projW[256,128] + projB (WMMA fp32)
  dim3 pgrid(NGRAPH / 16, 1);
  gemm_wmma_f32<<<pgrid, 256, 0, stream>>>(repr, projW, projB, (float*)d_out,
                                           NGRAPH, HIDDIM, 2 * HIDDIM);
}